// LawModel_62130996904084
// MI455X (gfx1250) — compile-verified
//
#include <hip/hip_runtime.h>
#include <hip/hip_bf16.h>
#include <math.h>

#define H2 256            // node feature width (2*hidden)
#define WAVE 32

typedef __attribute__((ext_vector_type(2))) float v2f;
typedef __attribute__((ext_vector_type(8))) float v8f;

// D = A(16x4 f32) * B(4x16 f32) + C(16x16 f32), full-precision CDNA5 WMMA.
__device__ __forceinline__ v8f wmma_f32_4(v2f a, v2f b, v8f c) {
    return __builtin_amdgcn_wmma_f32_16x16x4_f32(
        /*neg_a=*/false, a, /*neg_b=*/false, b,
        /*c_mod=*/(short)0, c, /*reuse_a=*/false, /*reuse_b=*/false);
}

// ---------------------------------------------------------------------------
// CSR construction: degree count -> exclusive scan -> bucket fill.
// Only cheap int atomics (640K total) instead of 82M fp32 atomic RMWs.
// ---------------------------------------------------------------------------
__global__ void zero_i32_kernel(int* __restrict__ p, int n) {
    int i = blockIdx.x * blockDim.x + threadIdx.x;
    if (i < n) p[i] = 0;
}

__global__ void degree_count_kernel(const int* __restrict__ edst,
                                    int* __restrict__ deg, int n_edges) {
    int e = blockIdx.x * blockDim.x + threadIdx.x;
    if (e < n_edges) atomicAdd(&deg[edst[e]], 1);
}

// Single-block exclusive scan (n ~ 10000): 256-wide Hillis-Steele chunks + carry.
__global__ void scan_kernel(const int* __restrict__ deg,
                            int* __restrict__ offs,
                            int* __restrict__ cursor, int n) {
    __shared__ int buf[256];
    __shared__ int carry_s;
    if (threadIdx.x == 0) carry_s = 0;
    __syncthreads();
    for (int base = 0; base < n; base += 256) {
        const int i = base + (int)threadIdx.x;
        const int v = (i < n) ? deg[i] : 0;
        const int c = carry_s;
        buf[threadIdx.x] = v;
        __syncthreads();
        for (int off = 1; off < 256; off <<= 1) {
            int t = (threadIdx.x >= (unsigned)off) ? buf[threadIdx.x - off] : 0;
            __syncthreads();
            buf[threadIdx.x] += t;
            __syncthreads();
        }
        const int excl = buf[threadIdx.x] - v;
        if (i < n) { offs[i] = c + excl; cursor[i] = c + excl; }
        const int tot = buf[255];
        __syncthreads();
        if (threadIdx.x == 0) carry_s = c + tot;
        __syncthreads();
    }
    if (threadIdx.x == 0) offs[n] = carry_s;
}

__global__ void csr_fill_kernel(const int* __restrict__ esrc,
                                const int* __restrict__ edst,
                                int* __restrict__ cursor,
                                int* __restrict__ csr_src, int n_edges) {
    int e = blockIdx.x * blockDim.x + threadIdx.x;
    if (e < n_edges) {
        int pos = atomicAdd(&cursor[edst[e]], 1);
        csr_src[pos] = esrc[e];
    }
}

// ---------------------------------------------------------------------------
// Gather: one wave per node.  pooled[i] = deg(i)*h_i - sum_j h_{src(j)}
// Each lane owns 8 contiguous floats (2x float4). Pure L2-resident reads,
// register accumulation, one float4-pair store. No float atomics at all.
// ---------------------------------------------------------------------------
__global__ void node_gather_kernel(const float* __restrict__ nf,
                                   const int* __restrict__ offs,
                                   const int* __restrict__ csr_src,
                                   float* __restrict__ pooled, int n_nodes) {
    const int wid  = (blockIdx.x * blockDim.x + threadIdx.x) >> 5;
    const int lane = threadIdx.x & (WAVE - 1);
    if (wid >= n_nodes) return;

    const int beg = offs[wid];
    const int end = offs[wid + 1];
    const int b4  = lane * 2;                 // float4 index; 32 lanes * 8 floats

    float4 acc0 = make_float4(0.f, 0.f, 0.f, 0.f);
    float4 acc1 = make_float4(0.f, 0.f, 0.f, 0.f);
    for (int p = beg; p < end; ++p) {
        const int s = csr_src[p];
        const float4* __restrict__ ps = (const float4*)(nf + (size_t)s * H2);
        float4 v0 = ps[b4];
        float4 v1 = ps[b4 + 1];
        acc0.x += v0.x; acc0.y += v0.y; acc0.z += v0.z; acc0.w += v0.w;
        acc1.x += v1.x; acc1.y += v1.y; acc1.z += v1.z; acc1.w += v1.w;
    }
    const float degf = (float)(end - beg);
    const float4* __restrict__ pd = (const float4*)(nf + (size_t)wid * H2);
    float4 d0 = pd[b4];
    float4 d1 = pd[b4 + 1];
    float4 r0, r1;
    r0.x = degf * d0.x - acc0.x; r0.y = degf * d0.y - acc0.y;
    r0.z = degf * d0.z - acc0.z; r0.w = degf * d0.w - acc0.w;
    r1.x = degf * d1.x - acc1.x; r1.y = degf * d1.y - acc1.y;
    r1.z = degf * d1.z - acc1.z; r1.w = degf * d1.w - acc1.w;
    float4* __restrict__ pp = (float4*)(pooled + (size_t)wid * H2);
    pp[b4]     = r0;
    pp[b4 + 1] = r1;
}

// ---------------------------------------------------------------------------
// GEMM1: h = [node_feats | pooled] @ W1 + b1  (K=512, fp32 WMMA).
// Block = 8 M-tiles x 1 N-tile. W1[:,16] slab (32KB) staged once in LDS,
// pair-interleaved so each lane's (k,k+1) B pair is a single ds_load_b64.
// Inner loop: 1 global b64 (A) + 1 ds b64 (B) + 1 v_wmma.
// ---------------------------------------------------------------------------
__global__ void __launch_bounds__(256)
gemm1_kernel(const float* __restrict__ nf,
             const float* __restrict__ pooled,
             const float* __restrict__ W1,
             const float* __restrict__ b1,
             float* __restrict__ h, int n_rows) {
    __shared__ float Bs[2 * H2 * 16];                 // 8192 floats = 32 KB
    const int nTile   = blockIdx.y;
    const int colBase = nTile * 16;

    for (int id = threadIdx.x; id < 2 * H2 * 16; id += 256) {
        const int k = id >> 4;
        const int c = id & 15;
        Bs[((k >> 1) << 5) + (c << 1) + (k & 1)] = W1[(size_t)k * H2 + colBase + c];
    }
    __syncthreads();

    const int wave = threadIdx.x >> 5;
    const int lane = threadIdx.x & (WAVE - 1);
    const int half = lane >> 4;
    const int l16  = lane & 15;
    const int mTile = blockIdx.x * 8 + wave;          // wave-uniform
    if (mTile >= n_rows / 16) return;                 // EXEC stays all-1s per wave

    const int row = mTile * 16 + l16;
    const int col = colBase + l16;
    const float* __restrict__ a0 = nf     + (size_t)row * H2;  // K 0..255
    const float* __restrict__ a1 = pooled + (size_t)row * H2;  // K 256..511
    const int ka = 2 * half;

    v8f acc = {};
#pragma unroll 4
    for (int k = 0; k < H2; k += 4) {
        v2f a = *(const v2f*)(a0 + k + ka);
        const int kp = (k + ka) >> 1;
        v2f b = *(const v2f*)(&Bs[(kp << 5) + (l16 << 1)]);
        acc = wmma_f32_4(a, b, acc);
    }
#pragma unroll 4
    for (int k = 0; k < H2; k += 4) {
        v2f a = *(const v2f*)(a1 + k + ka);
        const int kp = (H2 + k + ka) >> 1;
        v2f b = *(const v2f*)(&Bs[(kp << 5) + (l16 << 1)]);
        acc = wmma_f32_4(a, b, acc);
    }

    const float bias = b1[col];
#pragma unroll
    for (int v = 0; v < 8; ++v) {
        const int m = mTile * 16 + v + 8 * half;      // C/D: M = v + 8*half
        h[(size_t)m * H2 + col] = acc[v] + bias;
    }
}

// ---------------------------------------------------------------------------
// GEMM2: out = tanh(h @ W2 + b2)  (K=256, fp32 WMMA). Same LDS-staged scheme.
// ---------------------------------------------------------------------------
__global__ void __launch_bounds__(256)
gemm2_kernel(const float* __restrict__ hbuf,
             const float* __restrict__ W2,
             const float* __restrict__ b2,
             float* __restrict__ out, int n_rows) {
    __shared__ float Bs[H2 * 16];                     // 4096 floats = 16 KB
    const int nTile   = blockIdx.y;
    const int colBase = nTile * 16;

    for (int id = threadIdx.x; id < H2 * 16; id += 256) {
        const int k = id >> 4;
        const int c = id & 15;
        Bs[((k >> 1) << 5) + (c << 1) + (k & 1)] = W2[(size_t)k * H2 + colBase + c];
    }
    __syncthreads();

    const int wave = threadIdx.x >> 5;
    const int lane = threadIdx.x & (WAVE - 1);
    const int half = lane >> 4;
    const int l16  = lane & 15;
    const int mTile = blockIdx.x * 8 + wave;
    if (mTile >= n_rows / 16) return;

    const int row = mTile * 16 + l16;
    const int col = colBase + l16;
    const float* __restrict__ arow = hbuf + (size_t)row * H2;
    const int ka = 2 * half;

    v8f acc = {};
#pragma unroll 4
    for (int k = 0; k < H2; k += 4) {
        v2f a = *(const v2f*)(arow + k + ka);
        const int kp = (k + ka) >> 1;
        v2f b = *(const v2f*)(&Bs[(kp << 5) + (l16 << 1)]);
        acc = wmma_f32_4(a, b, acc);
    }

    const float bias = b2[col];
#pragma unroll
    for (int v = 0; v < 8; ++v) {
        const int m = mTile * 16 + v + 8 * half;
        out[(size_t)m * H2 + col] = tanhf(acc[v] + bias);
    }
}

// ---------------------------------------------------------------------------
// Launch: inputs are {node_feats, W1, b1, W2, b2, edge_src, edge_dst}
// ---------------------------------------------------------------------------
extern "C" void kernel_launch(void* const* d_in, const int* in_sizes, int n_in,
                              void* d_out, int out_size, void* d_ws, size_t ws_size,
                              hipStream_t stream) {
    const float* nf   = (const float*)d_in[0];
    const float* W1   = (const float*)d_in[1];
    const float* b1   = (const float*)d_in[2];
    const float* W2   = (const float*)d_in[3];
    const float* b2   = (const float*)d_in[4];
    const int*   esrc = (const int*)d_in[5];
    const int*   edst = (const int*)d_in[6];

    const int n_nodes = in_sizes[0] / H2;   // 10000
    const int n_edges = in_sizes[5];        // 320000

    // Workspace carving (all chunks 16B-aligned by construction).
    char* ws = (char*)d_ws;
    float* pooled  = (float*)ws;                       ws += (size_t)n_nodes * H2 * 4;
    float* hbuf    = (float*)ws;                       ws += (size_t)n_nodes * H2 * 4;
    int*   deg     = (int*)ws;                         ws += (size_t)n_nodes * 4;
    int*   offs    = (int*)ws;                         ws += (size_t)(n_nodes + 4) * 4;
    int*   cursor  = (int*)ws;                         ws += (size_t)n_nodes * 4;
    int*   csr_src = (int*)ws;

    // 1) CSR build: zero degrees, count, scan, fill
    zero_i32_kernel<<<(n_nodes + 255) / 256, 256, 0, stream>>>(deg, n_nodes);
    degree_count_kernel<<<(n_edges + 255) / 256, 256, 0, stream>>>(edst, deg, n_edges);
    scan_kernel<<<1, 256, 0, stream>>>(deg, offs, cursor, n_nodes);
    csr_fill_kernel<<<(n_edges + 255) / 256, 256, 0, stream>>>(esrc, edst, cursor, csr_src, n_edges);

    // 2) Per-node gather (one wave32 per node, 8 waves/block)
    node_gather_kernel<<<(n_nodes + 7) / 8, 256, 0, stream>>>(nf, offs, csr_src, pooled, n_nodes);

    // 3) GEMM1 (K=512) into h: grid = (M-tile groups) x (N tiles)
    {
        const int mTiles = n_nodes / 16;               // 625
        dim3 grid((mTiles + 7) / 8, H2 / 16);
        gemm1_kernel<<<grid, 256, 0, stream>>>(nf, pooled, W1, b1, hbuf, n_nodes);
    }
    // 4) GEMM2 (K=256) + tanh into d_out
    {
        const int mTiles = n_nodes / 16;
        dim3 grid((mTiles + 7) / 8, H2 / 16);
        gemm2_kernel<<<grid, 256, 0, stream>>>(hbuf, W2, b2, (float*)d_out, n_nodes);
    }
}